// NNEncoder_74844100100208
// MI455X (gfx1250) — compile-verified
//
#include <hip/hip_runtime.h>

typedef __attribute__((ext_vector_type(2))) float v2f;
typedef __attribute__((ext_vector_type(8))) float v8f;

#define BN_EPS 1e-5f
#define KF 128           // in_feats (specialized)
#define NT 7             // N tiles of 16 -> 112 padded outputs
#define NPAD 112
#define KCHUNKS 32       // 128 / 4

// ---------------- kernel: zero a float region ----------------
__global__ void k_zero(float* __restrict__ p, long n) {
    long i = (long)blockIdx.x * blockDim.x + threadIdx.x;
    long stride = (long)gridDim.x * blockDim.x;
    for (; i < n; i += stride) p[i] = 0.0f;
}

// ---------------- kernel: per-feature sum & sum-of-squares ----------------
// 256 threads: thread t handles feature t&127, rows 2*blk + (t>>7) stride 2*grid.
__global__ void k_stats(const float* __restrict__ e, long n_edges,
                        float* __restrict__ stats /* [0,128)=sum, [128,256)=sumsq */) {
    __shared__ float ssum[256];
    __shared__ float ssq[256];
    const int tid  = threadIdx.x;
    const int c    = tid & 127;
    const int half = tid >> 7;
    float s = 0.0f, q = 0.0f;
    long r = (long)blockIdx.x * 2 + half;
    const long stride = (long)gridDim.x * 2;
    for (; r < n_edges; r += stride) {
        float v = e[r * KF + c];
        s += v;
        q += v * v;
    }
    ssum[tid] = s;
    ssq[tid]  = q;
    __syncthreads();
    if (tid < 128) {
        atomicAdd(&stats[c],       ssum[tid] + ssum[tid + 128]);
        atomicAdd(&stats[128 + c], ssq[tid]  + ssq[tid + 128]);
    }
}

// ---------------- kernel: fold BN into W,b -> W' (112x128, zero-padded), b' (112) ----------
__global__ void k_finalize(const float* __restrict__ stats,
                           const float* __restrict__ gamma,
                           const float* __restrict__ beta,
                           const float* __restrict__ W,
                           const float* __restrict__ b,
                           float n_inv, int out_f,
                           float* __restrict__ Wp, float* __restrict__ bp) {
    __shared__ float scale_s[KF];
    __shared__ float shift_s[KF];
    __shared__ float red[KF];
    const int i = threadIdx.x;   // 128 threads
    float mean = stats[i] * n_inv;
    float var  = stats[KF + i] * n_inv - mean * mean;
    float sc   = gamma[i] * rsqrtf(var + BN_EPS);
    float sh   = beta[i] - mean * sc;
    scale_s[i] = sc;
    shift_s[i] = sh;
    __syncthreads();
    for (int o = 0; o < NPAD; ++o) {
        float w = (o < out_f) ? W[(long)o * KF + i] : 0.0f;
        Wp[(long)o * KF + i] = w * scale_s[i];
        red[i] = w * shift_s[i];
        __syncthreads();
        for (int off = 64; off > 0; off >>= 1) {
            if (i < off) red[i] += red[i + off];
            __syncthreads();
        }
        if (i == 0) bp[o] = ((o < out_f) ? b[o] : 0.0f) + red[0];
        __syncthreads();
    }
}

// ---------------- kernel: fp32 WMMA GEMM + bias + ReLU + scatter-add ----------------
// Block = 128 threads = 4 waves; each wave owns a 16-edge strip (block covers 64 edges).
// A (16x4 f32) per ISA layout: lane m=(l&15) is row M, hi=(l>>4) selects K pair {2hi,2hi+1}.
// B fragments for all 7 N-tiles are staged into LDS in exact per-lane order
// (index = ((nt*32+c)*32+lane)) so each ds_load_b64 is 256B contiguous, conflict-free.
__global__ void k_gemm_scatter(const float* __restrict__ e,
                               const int*   __restrict__ dst,
                               const float* __restrict__ Wp,
                               const float* __restrict__ bp,
                               float* __restrict__ sums,
                               float* __restrict__ cnt,
                               long n_edges, int out_f) {
    __shared__ v2f ldsB[NT * KCHUNKS * 32];   // 7*32*32*8B = 57344 B

    const int tid  = threadIdx.x;
    const int lane = tid & 31;
    const int wave = tid >> 5;
    const int m    = lane & 15;
    const int hi   = lane >> 4;

    // ---- stage swizzled B (W'^T fragments) into LDS, once per block ----
    for (int fi = tid; fi < NT * KCHUNKS * 32; fi += 128) {
        int l  = fi & 31;
        int c  = (fi >> 5) & 31;
        int nt = fi >> 10;
        int n  = nt * 16 + (l & 15);
        int k  = 4 * c + 2 * (l >> 4);
        ldsB[fi] = *(const v2f*)(Wp + (long)n * KF + k);
    }
    __syncthreads();

    const long edgeBase = (long)blockIdx.x * 64 + (long)wave * 16;

    // ---- load full 16x128 A strip into registers (32 chunks x v2f) ----
    long rowLd = edgeBase + m;
    if (rowLd >= n_edges) rowLd = n_edges - 1;     // clamp (keeps EXEC full)
    const float* erow = e + rowLd * KF + 2 * hi;
    v2f a[KCHUNKS];
#pragma unroll
    for (int c = 0; c < KCHUNKS; ++c)
        a[c] = *(const v2f*)(erow + 4 * c);

    // ---- preload dst ids for my 8 accumulator rows ----
    int  dloc[8];
    bool eval[8];
#pragma unroll
    for (int v = 0; v < 8; ++v) {
        long edge = edgeBase + v + 8 * hi;
        eval[v] = (edge < n_edges);
        dloc[v] = eval[v] ? dst[edge] : 0;
    }

    // ---- GEMM over 7 N-tiles, K=128 as 32 x (16x16x4 f32 WMMA) ----
    for (int nt = 0; nt < NT; ++nt) {
        v8f acc = {};
        const v2f* bfr = ldsB + nt * (KCHUNKS * 32) + lane;
#pragma unroll
        for (int c = 0; c < KCHUNKS; ++c) {
            v2f bv = bfr[c * 32];
            acc = __builtin_amdgcn_wmma_f32_16x16x4_f32(
                /*neg_a=*/false, a[c], /*neg_b=*/false, bv,
                /*c_mod=*/(short)0, acc, /*reuse_a=*/false, /*reuse_b=*/false);
        }
        // epilogue: bias + ReLU + scatter-add (lanes 0-15 hit 16 consecutive
        // floats of one node row -> coalesced L2 atomics)
        const int out = nt * 16 + m;
        if (out < out_f) {
            const float bias = bp[out];
#pragma unroll
            for (int v = 0; v < 8; ++v) {
                if (eval[v]) {
                    float val = acc[v] + bias;
                    val = val > 0.0f ? val : 0.0f;
                    atomicAdd(&sums[(long)dloc[v] * out_f + out], val);
                }
            }
        }
    }

    // ---- per-edge count (each edge counted exactly once: lanes with m==0) ----
    if (m == 0) {
#pragma unroll
        for (int v = 0; v < 8; ++v)
            if (eval[v]) atomicAdd(&cnt[dloc[v]], 1.0f);
    }
}

// ---------------- kernel: h = sums / max(cnt,1) ----------------
__global__ void k_mean(const float* __restrict__ sums,
                       const float* __restrict__ cnt,
                       float* __restrict__ out, long total, int out_f) {
    long i = (long)blockIdx.x * blockDim.x + threadIdx.x;
    long stride = (long)gridDim.x * blockDim.x;
    for (; i < total; i += stride) {
        long node = i / out_f;
        out[i] = sums[i] / fmaxf(cnt[node], 1.0f);
    }
}

extern "C" void kernel_launch(void* const* d_in, const int* in_sizes, int n_in,
                              void* d_out, int out_size, void* d_ws, size_t ws_size,
                              hipStream_t stream) {
    const float* e     = (const float*)d_in[0];
    const int*   dst   = (const int*)  d_in[1];
    const float* gamma = (const float*)d_in[2];
    const float* beta  = (const float*)d_in[3];
    const float* W     = (const float*)d_in[4];
    const float* b     = (const float*)d_in[5];

    const int  in_f    = in_sizes[2];                 // 128
    const int  out_f   = in_sizes[5];                 // 100
    const long n_edges = (long)in_sizes[0] / in_f;    // 1.6M
    const int  n_nodes = out_size / out_f;            // 50000

    // ---- workspace layout (floats) ----
    float* ws = (float*)d_ws;
    long off = 0;
    float* stats = ws + off; off += 256;
    float* cnt   = ws + off; off += ((long)n_nodes + 3) & ~3L;
    float* bp    = ws + off; off += NPAD;
    float* Wp    = ws + off; off += (long)NPAD * KF;   // 16B-aligned (off multiple of 4)
    float* sums  = ws + off; off += (long)n_nodes * out_f;
    const long zeroN = off;   // zero everything (bp/Wp are fully overwritten anyway)

    k_zero<<<2048, 256, 0, stream>>>(ws, zeroN);
    k_stats<<<1024, 256, 0, stream>>>(e, n_edges, stats);
    k_finalize<<<1, 128, 0, stream>>>(stats, gamma, beta, W, b,
                                      (float)(1.0 / (double)n_edges), out_f, Wp, bp);
    const long ntiles = (n_edges + 15) / 16;
    const int  blocks = (int)((ntiles + 3) / 4);
    k_gemm_scatter<<<blocks, 128, 0, stream>>>(e, dst, Wp, bp, sums, cnt, n_edges, out_f);
    k_mean<<<(out_size + 255) / 256, 256, 0, stream>>>(sums, cnt, (float*)d_out,
                                                       (long)out_size, out_f);
}